// S4Base_86741159510116
// MI455X (gfx1250) — compile-verified
//
#include <hip/hip_runtime.h>

#define L_SEQ 4096
#define D_DIM 1024
#define N_ST  16
#define T_CH  16
#define CH    (L_SEQ / T_CH)   // 256 chunks
#define CG    16               // chunks per wave tile
#define NCG   (CH / CG)        // 16 tile groups
#define DB    32               // channels per block (32 waves)

typedef __attribute__((ext_vector_type(2))) float v2f;
typedef __attribute__((ext_vector_type(8))) float v8f;

__device__ __forceinline__ v8f wmma_f32(v2f a, v2f b, v8f c) {
    // V_WMMA_F32_16X16X4_F32 : D(16x16 f32) = A(16x4) * B(4x16) + C
    return __builtin_amdgcn_wmma_f32_16x16x4_f32(
        false, a, false, b, (short)0, c, false, false);
}

// ---------------------------------------------------------------------------
// Kernel 1: per-(d,n) parameter tables.
//   Ew[d][n][k]   = Lam^(15-k) * dtB        (complex, re/im arrays)
//   Wt[d][t][n]   = C * Lam^(t+1)           (re and NEGATED im arrays)
//   LamT[d*16+n]  = Lam^16
// ---------------------------------------------------------------------------
__global__ void k_params(const float* __restrict__ Delta,
                         const float* __restrict__ A_re, const float* __restrict__ A_im,
                         const float* __restrict__ B_re, const float* __restrict__ B_im,
                         const float* __restrict__ C_re, const float* __restrict__ C_im,
                         float* __restrict__ Ew_re, float* __restrict__ Ew_im,
                         float* __restrict__ Wt_re, float* __restrict__ Wt_mi,
                         float* __restrict__ LamT_re, float* __restrict__ LamT_im) {
    int tid = blockIdx.x * blockDim.x + threadIdx.x;   // tid = d*16 + n
    int d = tid >> 4, n = tid & 15;
    float del = Delta[d];
    // stable softplus
    float dt = fmaxf(del, 0.0f) + log1pf(expf(-fabsf(del)));
    float ar = dt * A_re[tid], ai = dt * A_im[tid];
    float sn, cs;
    sincosf(ai, &sn, &cs);
    float er = expf(ar);
    float Lr = er * cs, Li = er * sn;              // Lam
    float br = dt * B_re[tid], bi = dt * B_im[tid]; // dtB
    float cr = C_re[tid], ci = C_im[tid];
    // Lam^16 directly via exp (accurate)
    float s16, c16;
    sincosf(16.0f * ai, &s16, &c16);
    float e16 = expf(16.0f * ar);
    LamT_re[tid] = e16 * c16;
    LamT_im[tid] = e16 * s16;
    // iterate powers Lam^j
    float pr = 1.0f, pi = 0.0f;
    #pragma unroll
    for (int j = 0; j < 16; ++j) {
        // E[n][k=15-j] = Lam^j * dtB
        float exr = pr * br - pi * bi;
        float exi = pr * bi + pi * br;
        Ew_re[tid * 16 + (15 - j)] = exr;
        Ew_im[tid * 16 + (15 - j)] = exi;
        // advance to Lam^(j+1)
        float nr = pr * Lr - pi * Li;
        float ni = pr * Li + pi * Lr;
        pr = nr; pi = ni;
        // W[t=j][n] = C * Lam^(j+1)
        float wr = cr * pr - ci * pi;
        float wi = cr * pi + ci * pr;
        Wt_re[(d * 16 + j) * 16 + n] = wr;
        Wt_mi[(d * 16 + j) * 16 + n] = -wi;   // store negated imag (f32 WMMA has no B-neg)
    }
}

// ---------------------------------------------------------------------------
// Kernel 2: Toeplitz filter taps g[d][j] = Re( sum_n C * Lam^j * dtB )
// ---------------------------------------------------------------------------
__global__ void k_gtaps(const float* __restrict__ C_re, const float* __restrict__ C_im,
                        const float* __restrict__ Ew_re, const float* __restrict__ Ew_im,
                        float* __restrict__ g) {
    int tid = blockIdx.x * blockDim.x + threadIdx.x;   // tid = d*16 + j
    int d = tid >> 4, j = tid & 15;
    float acc = 0.0f;
    #pragma unroll
    for (int n = 0; n < 16; ++n) {
        float cr = C_re[d * 16 + n], ci = C_im[d * 16 + n];
        float er = Ew_re[(d * 16 + n) * 16 + (15 - j)];  // = Re(Lam^j dtB)
        float ei = Ew_im[(d * 16 + n) * 16 + (15 - j)];
        acc += cr * er - ci * ei;
    }
    g[tid] = acc;
}

// ---------------------------------------------------------------------------
// Kernel 3: transpose x (L,D) -> xt (D,L), LDS tiled
// ---------------------------------------------------------------------------
__global__ void k_transpose(const float* __restrict__ x, float* __restrict__ xt) {
    __shared__ float t[32][33];
    int l0 = blockIdx.x * 32, d0 = blockIdx.y * 32;
    int tx = threadIdx.x, ty = threadIdx.y;   // (32, 8)
    #pragma unroll
    for (int i = 0; i < 4; ++i)
        t[ty + i * 8][tx] = x[(size_t)(l0 + ty + i * 8) * D_DIM + d0 + tx];
    __syncthreads();
    #pragma unroll
    for (int i = 0; i < 4; ++i)
        xt[(size_t)(d0 + ty + i * 8) * L_SEQ + l0 + tx] = t[tx][ty + i * 8];
}

// ---------------------------------------------------------------------------
// Kernel 4 (WMMA pass A): chunk end states.
//   b_end^T (16c x 16n) = X^T (16c x 16k) @ E^T (16k x 16n), per channel d.
//   Output layout bre/bim[c][d][n] (coalesced for the carry scan).
// ---------------------------------------------------------------------------
__global__ __launch_bounds__(1024) void k_passA(
        const float* __restrict__ xt,
        const float* __restrict__ Ew_re, const float* __restrict__ Ew_im,
        float* __restrict__ bre, float* __restrict__ bim) {
    __shared__ float st[CG * DB * N_ST];   // 16*32*16 floats = 32 KB
    int w = threadIdx.x >> 5, lane = threadIdx.x & 31;
    int d = blockIdx.x * DB + w;
    int c0 = blockIdx.y * CG;
    int cs = lane & 15, hi = lane >> 4;

    // A fragments (X^T): lane c=cs holds x[l=16*(c0+cs)+k+2*hi + {0,1}] of channel d
    const float* xrow = xt + (size_t)d * L_SEQ + 16 * (size_t)(c0 + cs) + 2 * hi;
    v2f a0 = *(const v2f*)(xrow + 0);
    v2f a1 = *(const v2f*)(xrow + 4);
    v2f a2 = *(const v2f*)(xrow + 8);
    v2f a3 = *(const v2f*)(xrow + 12);
    // B fragments (E^T): lane n=cs holds E[n][k+2*hi + {0,1}]
    const float* err = Ew_re + (d * 16 + cs) * 16 + 2 * hi;
    const float* eir = Ew_im + (d * 16 + cs) * 16 + 2 * hi;

    v8f accr = {}; v8f acci = {};
    accr = wmma_f32(a0, *(const v2f*)(err + 0),  accr);
    accr = wmma_f32(a1, *(const v2f*)(err + 4),  accr);
    accr = wmma_f32(a2, *(const v2f*)(err + 8),  accr);
    accr = wmma_f32(a3, *(const v2f*)(err + 12), accr);
    acci = wmma_f32(a0, *(const v2f*)(eir + 0),  acci);
    acci = wmma_f32(a1, *(const v2f*)(eir + 4),  acci);
    acci = wmma_f32(a2, *(const v2f*)(eir + 8),  acci);
    acci = wmma_f32(a3, *(const v2f*)(eir + 12), acci);

    size_t base = (size_t)c0 * (D_DIM * N_ST) + (size_t)blockIdx.x * DB * N_ST;
    // stage+store real part: D frag: VGPR r, lane -> value at (c=r+8*hi, n=cs)
    #pragma unroll
    for (int r = 0; r < 8; ++r)
        st[(r + 8 * hi) * (DB * N_ST) + w * 16 + cs] = accr[r];
    __syncthreads();
    #pragma unroll
    for (int i = 0; i < 8; ++i) {
        int idx = threadIdx.x + i * 1024;          // [c][dsub][n] linear
        int c = idx >> 9, rem = idx & 511;
        bre[base + (size_t)c * (D_DIM * N_ST) + rem] = st[idx];
    }
    __syncthreads();
    #pragma unroll
    for (int r = 0; r < 8; ++r)
        st[(r + 8 * hi) * (DB * N_ST) + w * 16 + cs] = acci[r];
    __syncthreads();
    #pragma unroll
    for (int i = 0; i < 8; ++i) {
        int idx = threadIdx.x + i * 1024;
        int c = idx >> 9, rem = idx & 511;
        bim[base + (size_t)c * (D_DIM * N_ST) + rem] = st[idx];
    }
}

// ---------------------------------------------------------------------------
// Kernel 5: inter-chunk carry scan, in place: b[c] <- h0[c] (carry before chunk c)
//   carry' = Lam^16 * carry + b_end[c]
// ---------------------------------------------------------------------------
__global__ void k_scan(float* __restrict__ bre, float* __restrict__ bim,
                       const float* __restrict__ LamT_re, const float* __restrict__ LamT_im) {
    int tid = blockIdx.x * blockDim.x + threadIdx.x;   // tid = d*16+n
    float LTr = LamT_re[tid], LTi = LamT_im[tid];
    float cr = 0.0f, ci = 0.0f;
    for (int c = 0; c < CH; ++c) {
        size_t o = (size_t)c * (D_DIM * N_ST) + tid;
        float tr = bre[o], ti = bim[o];
        bre[o] = cr; bim[o] = ci;
        float nr = LTr * cr - LTi * ci + tr;
        float ni = LTr * ci + LTi * cr + ti;
        cr = nr; ci = ni;
    }
}

// ---------------------------------------------------------------------------
// Kernel 6 (WMMA pass B): final output.
//   O^T (16c x 16t) = X^T@G^T + H0re^T@Wre^T + H0im^T@(-Wim)^T  per channel d
//   then out[l,d] = O + D_param[d]*x[l,d], coalesced via LDS restage.
// ---------------------------------------------------------------------------
__global__ __launch_bounds__(1024) void k_passB(
        const float* __restrict__ xt, const float* __restrict__ x,
        const float* __restrict__ g,
        const float* __restrict__ Wt_re, const float* __restrict__ Wt_mi,
        const float* __restrict__ bre, const float* __restrict__ bim,
        const float* __restrict__ Dp, float* __restrict__ out) {
    __shared__ float st[DB * 257];   // 32 rows (d) x 256 (+1 pad) l-values
    int w = threadIdx.x >> 5, lane = threadIdx.x & 31;
    int d = blockIdx.x * DB + w;
    int c0 = blockIdx.y * CG;
    int cs = lane & 15, hi = lane >> 4;

    // A fragments X^T
    const float* xrow = xt + (size_t)d * L_SEQ + 16 * (size_t)(c0 + cs) + 2 * hi;
    v2f ax0 = *(const v2f*)(xrow + 0);
    v2f ax1 = *(const v2f*)(xrow + 4);
    v2f ax2 = *(const v2f*)(xrow + 8);
    v2f ax3 = *(const v2f*)(xrow + 12);
    // A fragments H0^T: lane c=cs holds h0[n=k+2*hi+{0,1}] of chunk c0+cs
    const float* hr = bre + (size_t)(c0 + cs) * (D_DIM * N_ST) + d * 16 + 2 * hi;
    const float* hm = bim + (size_t)(c0 + cs) * (D_DIM * N_ST) + d * 16 + 2 * hi;
    v2f hr0 = *(const v2f*)(hr + 0);
    v2f hr1 = *(const v2f*)(hr + 4);
    v2f hr2 = *(const v2f*)(hr + 8);
    v2f hr3 = *(const v2f*)(hr + 12);
    v2f hm0 = *(const v2f*)(hm + 0);
    v2f hm1 = *(const v2f*)(hm + 4);
    v2f hm2 = *(const v2f*)(hm + 8);
    v2f hm3 = *(const v2f*)(hm + 12);
    // B fragments W^T: lane t=cs holds W[t][k+2*hi+{0,1}]
    const float* wr = Wt_re + (d * 16 + cs) * 16 + 2 * hi;
    const float* wm = Wt_mi + (d * 16 + cs) * 16 + 2 * hi;
    const float* gd = g + d * 16;

    v8f acc = {};
    // Toeplitz conv: B frag G^T[kk][t] = (t >= k+kk) ? g[t-k-kk] : 0
    #pragma unroll
    for (int ks = 0; ks < 4; ++ks) {
        int k = ks * 4;
        int j0 = cs - k - 2 * hi;
        v2f bg;
        bg[0] = (j0 >= 0) ? gd[j0] : 0.0f;
        bg[1] = (j0 >= 1) ? gd[j0 - 1] : 0.0f;
        v2f a = (ks == 0) ? ax0 : (ks == 1) ? ax1 : (ks == 2) ? ax2 : ax3;
        acc = wmma_f32(a, bg, acc);
    }
    // carry contribution
    acc = wmma_f32(hr0, *(const v2f*)(wr + 0),  acc);
    acc = wmma_f32(hr1, *(const v2f*)(wr + 4),  acc);
    acc = wmma_f32(hr2, *(const v2f*)(wr + 8),  acc);
    acc = wmma_f32(hr3, *(const v2f*)(wr + 12), acc);
    acc = wmma_f32(hm0, *(const v2f*)(wm + 0),  acc);
    acc = wmma_f32(hm1, *(const v2f*)(wm + 4),  acc);
    acc = wmma_f32(hm2, *(const v2f*)(wm + 8),  acc);
    acc = wmma_f32(hm3, *(const v2f*)(wm + 12), acc);

    // stage: value(r,lane) sits at l_local = 16*(c=r+8*hi) + (t=cs)
    #pragma unroll
    for (int r = 0; r < 8; ++r)
        st[w * 257 + 16 * (r + 8 * hi) + cs] = acc[r];
    __syncthreads();

    int l0 = c0 * 16;   // 256 l-values per block
    #pragma unroll
    for (int i = 0; i < 8; ++i) {
        int idx = threadIdx.x + i * 1024;   // 8192 = 256 l x 32 d
        int ll = idx >> 5, dsub = idx & 31;
        int dg = blockIdx.x * DB + dsub;
        size_t go = (size_t)(l0 + ll) * D_DIM + dg;
        out[go] = st[dsub * 257 + ll] + Dp[dg] * x[go];
    }
}

// ---------------------------------------------------------------------------
extern "C" void kernel_launch(void* const* d_in, const int* in_sizes, int n_in,
                              void* d_out, int out_size, void* d_ws, size_t ws_size,
                              hipStream_t stream) {
    const float* x     = (const float*)d_in[0];
    const float* Delta = (const float*)d_in[1];
    const float* A_re  = (const float*)d_in[2];
    const float* A_im  = (const float*)d_in[3];
    const float* B_re  = (const float*)d_in[4];
    const float* B_im  = (const float*)d_in[5];
    const float* C_re  = (const float*)d_in[6];
    const float* C_im  = (const float*)d_in[7];
    const float* Dp    = (const float*)d_in[8];
    float* out = (float*)d_out;

    float* ws      = (float*)d_ws;
    float* xt      = ws;                    // D*L            = 4194304
    float* bre     = ws + 4194304;          // CH*D*N         = 4194304
    float* bim     = ws + 8388608;          // CH*D*N         = 4194304
    float* Ew_re   = ws + 12582912;         // D*256
    float* Ew_im   = Ew_re + 262144;
    float* Wt_re   = Ew_im + 262144;
    float* Wt_mi   = Wt_re + 262144;
    float* g       = Wt_mi + 262144;        // D*16
    float* LamT_re = g + 16384;
    float* LamT_im = LamT_re + 16384;

    k_params<<<64, 256, 0, stream>>>(Delta, A_re, A_im, B_re, B_im, C_re, C_im,
                                     Ew_re, Ew_im, Wt_re, Wt_mi, LamT_re, LamT_im);
    k_gtaps<<<64, 256, 0, stream>>>(C_re, C_im, Ew_re, Ew_im, g);
    k_transpose<<<dim3(L_SEQ / 32, D_DIM / 32), dim3(32, 8), 0, stream>>>(x, xt);
    k_passA<<<dim3(D_DIM / DB, NCG), 1024, 0, stream>>>(xt, Ew_re, Ew_im, bre, bim);
    k_scan<<<64, 256, 0, stream>>>(bre, bim, LamT_re, LamT_im);
    k_passB<<<dim3(D_DIM / DB, NCG), 1024, 0, stream>>>(xt, x, g, Wt_re, Wt_mi,
                                                        bre, bim, Dp, out);
}